// MWT_49709951484263
// MI455X (gfx1250) — compile-verified
//
#include <hip/hip_runtime.h>
#include <math.h>

// ---------------- problem constants ----------------
#define BB    32
#define NN    8192
#define CC    16
#define CKD   64            // C*K
#define BI    (BB*CKD)      // 2048 columns in spectral GEMMs
#define MODES 64

typedef __attribute__((ext_vector_type(2))) float v2f;
typedef __attribute__((ext_vector_type(8))) float v8f;

struct Filt { float ec_s[8][4]; float ec_d[8][4]; float rc_e[8][4]; float rc_o[8][4]; };

// ---------------- fp32 WMMA helper (V_WMMA_F32_16X16X4_F32) ----------------
__device__ __forceinline__ v8f wmma4(v2f a, v2f b, v8f c) {
  // (neg_a, A, neg_b, B, c_mod, C, reuse_a, reuse_b)
  return __builtin_amdgcn_wmma_f32_16x16x4_f32(false, a, false, b, (short)0, c, false, false);
}

// ---------------- decomposition: xa=[x_even,x_odd] @ EC_D / EC_S ----------------
__global__ void k_decompose(const float* __restrict__ xin, float* __restrict__ dout,
                            float* __restrict__ sout, int Nh, Filt F) {
  int idx = blockIdx.x * blockDim.x + threadIdx.x;       // over BB*Nh*CC
  int total = BB * Nh * CC;
  if (idx >= total) return;
  int c = idx & 15;
  int t = (idx >> 4) % Nh;
  int b = idx / (Nh * CC);
  const float* pe = xin + (((size_t)b * (2 * Nh) + 2 * t) * CC + c) * 4;
  const float* po = xin + (((size_t)b * (2 * Nh) + 2 * t + 1) * CC + c) * 4;
  float v[8];
#pragma unroll
  for (int j = 0; j < 4; ++j) { v[j] = pe[j]; v[4 + j] = po[j]; }
  float* pd = dout + (size_t)idx * 4;
  float* ps = sout + (size_t)idx * 4;
#pragma unroll
  for (int kk = 0; kk < 4; ++kk) {
    float sd = 0.f, ss = 0.f;
#pragma unroll
    for (int j = 0; j < 8; ++j) { sd += v[j] * F.ec_d[j][kk]; ss += v[j] * F.ec_s[j][kk]; }
    pd[kk] = sd; ps[kk] = ss;
  }
}

// ---------------- twiddle tables ----------------
__global__ void k_twid_fwd(float* __restrict__ tc, float* __restrict__ ts,
                           int Nl, int l, int lpad) {
  int idx = blockIdx.x * blockDim.x + threadIdx.x;       // lpad*Nl
  if (idx >= lpad * Nl) return;
  int m = idx / Nl, n = idx % Nl;
  float c = 0.f, s = 0.f;
  if (m < l) {
    int r = (int)(((long long)m * (long long)n) % Nl);
    float ang = 6.283185307179586f * (float)r / (float)Nl;
    c = __cosf(ang); s = __sinf(ang);
  }
  tc[idx] = c; ts[idx] = s;
}

__global__ void k_twid_inv(float* __restrict__ ti, int Nl, int l, int K2) {
  int idx = blockIdx.x * blockDim.x + threadIdx.x;       // Nl*K2
  if (idx >= Nl * K2) return;
  int n = idx / K2, kq = idx % K2;
  int m = kq >> 1;
  float v = 0.f;
  if (m < l) {
    int r = (int)(((long long)m * (long long)n) % Nl);
    float ang = 6.283185307179586f * (float)r / (float)Nl;
    bool nyq = (2 * m == Nl);
    float coef = ((m == 0) || nyq ? 1.0f : 2.0f) / (float)Nl;
    if ((kq & 1) == 0) v = coef * __cosf(ang);
    else               v = nyq ? 0.f : -coef * __sinf(ang);
  }
  ti[idx] = v;
}

// ---------------- forward truncated DFT as WMMA GEMM ----------------
// Xre[m, col] = sum_n tc[m,n]*x[b,n,i] ; Xim = -sum_n ts[m,n]*x   (col=b*64+i)
__global__ void __launch_bounds__(128) k_dft_fwd_wmma(
    const float* __restrict__ xin, const float* __restrict__ tc, const float* __restrict__ ts,
    float* __restrict__ Fre, float* __restrict__ Fim, int Nl) {
  int lane = threadIdx.x & 31;
  int wav  = threadIdx.x >> 5;
  int tile = blockIdx.x * 4 + wav;
  int mtile = tile >> 7;            // tiles over 2048 cols = 128 n-tiles
  int ntile = tile & 127;
  int m0  = mtile * 16;
  int col = ntile * 16 + (lane & 15);
  int b = col >> 6, i = col & 63;
  int am = m0 + (lane & 15);
  int ak = (lane >> 4) * 2;
  const float* arow_c = tc + (size_t)am * Nl + ak;
  const float* arow_s = ts + (size_t)am * Nl + ak;
  const float* bcol   = xin + (size_t)b * Nl * 64 + (size_t)ak * 64 + i;
  v8f accR = {0,0,0,0,0,0,0,0}, accI = {0,0,0,0,0,0,0,0};
  for (int n0 = 0; n0 < Nl; n0 += 4) {
    v2f ac, an, bv;
    ac.x = arow_c[n0];     ac.y = arow_c[n0 + 1];
    an.x = -arow_s[n0];    an.y = -arow_s[n0 + 1];
    bv.x = bcol[(size_t)n0 * 64];  bv.y = bcol[(size_t)(n0 + 1) * 64];
    accR = wmma4(ac, bv, accR);
    accI = wmma4(an, bv, accI);
  }
  int rbase = (lane < 16) ? 0 : 8;
#pragma unroll
  for (int r = 0; r < 8; ++r) {
    int m = m0 + rbase + r;
    Fre[(size_t)m * BI + col] = accR[r];
    Fim[(size_t)m * BI + col] = accI[r];
  }
}

__global__ void k_dft_fwd_scalar(const float* __restrict__ xin, const float* __restrict__ tc,
                                 const float* __restrict__ ts, float* __restrict__ Fre,
                                 float* __restrict__ Fim, int Nl, int l) {
  int idx = blockIdx.x * blockDim.x + threadIdx.x;       // l*BI
  if (idx >= l * BI) return;
  int m = idx / BI, col = idx % BI;
  int b = col >> 6, i = col & 63;
  float sr = 0.f, si = 0.f;
  for (int n = 0; n < Nl; ++n) {
    float v = xin[((size_t)b * Nl + n) * 64 + i];
    sr += v * tc[(size_t)m * Nl + n];
    si -= v * ts[(size_t)m * Nl + n];
  }
  Fre[(size_t)m * BI + col] = sr;
  Fim[(size_t)m * BI + col] = si;
}

// ---------------- per-mode complex mix (always WMMA): ----------------
// Ud[m] = Fd[m]@wA[m] + Fs[m]@wB[m] ; Us[m] = Fd[m]@wC[m]   (complex, 32x64x64)
// weights pre-transposed to wT[m][i][o]
__global__ void __launch_bounds__(256) k_modemix_wmma(
    const float* __restrict__ Fdre, const float* __restrict__ Fdim,
    const float* __restrict__ Fsre, const float* __restrict__ Fsim,
    const float* __restrict__ wAre, const float* __restrict__ wAim,
    const float* __restrict__ wBre, const float* __restrict__ wBim,
    const float* __restrict__ wCre, const float* __restrict__ wCim,
    float* __restrict__ UdP, float* __restrict__ UsP) {
  int m = blockIdx.x;
  int lane = threadIdx.x & 31;
  int wav  = threadIdx.x >> 5;          // 8 waves: 2 b-tiles x 4 o-tiles
  int bt = wav >> 2;
  int ot = wav & 3;
  int brow = bt * 16 + (lane & 15);
  int ocol = ot * 16 + (lane & 15);
  int ak = (lane >> 4) * 2;
  size_t fbase = (size_t)m * BI + (size_t)brow * 64 + ak;
  size_t wbase = (size_t)m * 4096 + (size_t)ak * 64 + ocol;   // wT[m][i][o]
  v8f udr = {0,0,0,0,0,0,0,0}, udi = udr, usr = udr, usi = udr;
  for (int kk = 0; kk < 64; kk += 4) {
    v2f adr, adi, adin, afr, afi, afin;
    adr.x = Fdre[fbase + kk];  adr.y = Fdre[fbase + kk + 1];
    adi.x = Fdim[fbase + kk];  adi.y = Fdim[fbase + kk + 1];
    afr.x = Fsre[fbase + kk];  afr.y = Fsre[fbase + kk + 1];
    afi.x = Fsim[fbase + kk];  afi.y = Fsim[fbase + kk + 1];
    adin.x = -adi.x; adin.y = -adi.y;
    afin.x = -afi.x; afin.y = -afi.y;
    size_t w0 = wbase + (size_t)kk * 64;
    size_t w1 = w0 + 64;
    v2f bAr, bAi, bBr, bBi, bCr, bCi;
    bAr.x = wAre[w0]; bAr.y = wAre[w1];
    bAi.x = wAim[w0]; bAi.y = wAim[w1];
    bBr.x = wBre[w0]; bBr.y = wBre[w1];
    bBi.x = wBim[w0]; bBi.y = wBim[w1];
    bCr.x = wCre[w0]; bCr.y = wCre[w1];
    bCi.x = wCim[w0]; bCi.y = wCim[w1];
    udr = wmma4(adr,  bAr, udr);
    udr = wmma4(adin, bAi, udr);
    udr = wmma4(afr,  bBr, udr);
    udr = wmma4(afin, bBi, udr);
    udi = wmma4(adr,  bAi, udi);
    udi = wmma4(adi,  bAr, udi);
    udi = wmma4(afr,  bBi, udi);
    udi = wmma4(afi,  bBr, udi);
    usr = wmma4(adr,  bCr, usr);
    usr = wmma4(adin, bCi, usr);
    usi = wmma4(adr,  bCi, usi);
    usi = wmma4(adi,  bCr, usi);
  }
  int rbase = (lane < 16) ? 0 : 8;
#pragma unroll
  for (int r = 0; r < 8; ++r) {
    int b = bt * 16 + rbase + r;
    size_t col = (size_t)b * 64 + ocol;
    UdP[(size_t)(2 * m)     * BI + col] = udr[r];
    UdP[(size_t)(2 * m + 1) * BI + col] = udi[r];
    UsP[(size_t)(2 * m)     * BI + col] = usr[r];
    UsP[(size_t)(2 * m + 1) * BI + col] = usi[r];
  }
}

// ---------------- inverse truncated DFT as WMMA GEMM ----------------
// y[n, col] = sum_k ti[n,k] * pack[k, col] ; pack rows = (re,im) interleaved per mode
__global__ void __launch_bounds__(128) k_dft_inv_wmma(
    const float* __restrict__ pack, const float* __restrict__ ti,
    float* __restrict__ yout, int Nl, int K2) {
  int lane = threadIdx.x & 31;
  int wav  = threadIdx.x >> 5;
  int tile = blockIdx.x * 4 + wav;
  int mtile = tile >> 7;
  int ntile = tile & 127;
  int n0  = mtile * 16;
  int col = ntile * 16 + (lane & 15);
  int an = n0 + (lane & 15);
  int ak = (lane >> 4) * 2;
  const float* arow = ti + (size_t)an * K2 + ak;
  const float* bcol = pack + (size_t)ak * BI + col;
  v8f acc = {0,0,0,0,0,0,0,0};
  for (int k0 = 0; k0 < K2; k0 += 4) {
    v2f av, bv;
    av.x = arow[k0]; av.y = arow[k0 + 1];
    bv.x = bcol[(size_t)k0 * BI]; bv.y = bcol[(size_t)(k0 + 1) * BI];
    acc = wmma4(av, bv, acc);
  }
  int b = col >> 6, o = col & 63;
  int rbase = (lane < 16) ? 0 : 8;
#pragma unroll
  for (int r = 0; r < 8; ++r) {
    int n = n0 + rbase + r;
    yout[((size_t)b * Nl + n) * 64 + o] = acc[r];
  }
}

__global__ void k_dft_inv_scalar(const float* __restrict__ pack, const float* __restrict__ ti,
                                 float* __restrict__ yout, int Nl, int K2) {
  int idx = blockIdx.x * blockDim.x + threadIdx.x;       // Nl*BI
  if (idx >= Nl * BI) return;
  int n = idx / BI, col = idx % BI;
  float y = 0.f;
  for (int k = 0; k < K2; ++k)
    y += ti[(size_t)n * K2 + k] * pack[(size_t)k * BI + col];
  int b = col >> 6, o = col & 63;
  yout[((size_t)b * Nl + n) * 64 + o] = y;
}

// ---------------- weight transpose: w[i][o][m] -> wT[m][i][o] ----------------
__global__ void k_transpose_w(const float* __restrict__ w, float* __restrict__ wT) {
  int idx = blockIdx.x * blockDim.x + threadIdx.x;       // 64*64*64
  if (idx >= 64 * 64 * 64) return;
  int o = idx & 63;
  int i = (idx >> 6) & 63;
  int m = idx >> 12;
  wT[idx] = w[((size_t)i * 64 + o) * 64 + m];
}

// ---------------- bottom transform: x @ t0_w.T + t0_b ----------------
__global__ void k_t0(const float* __restrict__ xin, const float* __restrict__ t0w,
                     const float* __restrict__ t0b, float* __restrict__ xout) {
  int idx = blockIdx.x * blockDim.x + threadIdx.x;       // BB*CC*4
  if (idx >= BB * CC * 4) return;
  int kk = idx & 3;
  int bc = idx >> 2;
  const float* px = xin + (size_t)bc * 4;
  float s = t0b[kk];
#pragma unroll
  for (int j = 0; j < 4; ++j) s += px[j] * t0w[kk * 4 + j];
  xout[idx] = s;
}

// ---------------- reconstruction: [x+Us, Ud] @ RC_E/RC_O, interleave ----------------
__global__ void k_recon(const float* __restrict__ xin, const float* __restrict__ us,
                        const float* __restrict__ ud, float* __restrict__ xout, int M, Filt F) {
  int idx = blockIdx.x * blockDim.x + threadIdx.x;       // BB*M*CC
  int total = BB * M * CC;
  if (idx >= total) return;
  int c = idx & 15;
  int t = (idx >> 4) % M;
  int b = idx / (M * CC);
  const float* px = xin + (size_t)idx * 4;
  const float* pu = us + (size_t)idx * 4;
  const float* pd = ud + (size_t)idx * 4;
  float v[8];
#pragma unroll
  for (int j = 0; j < 4; ++j) { v[j] = px[j] + pu[j]; v[4 + j] = pd[j]; }
  float* pe = xout + (((size_t)b * (2 * M) + 2 * t) * CC + c) * 4;
  float* po = xout + (((size_t)b * (2 * M) + 2 * t + 1) * CC + c) * 4;
#pragma unroll
  for (int kk = 0; kk < 4; ++kk) {
    float e = 0.f, oo = 0.f;
#pragma unroll
    for (int j = 0; j < 8; ++j) { e += v[j] * F.rc_e[j][kk]; oo += v[j] * F.rc_o[j][kk]; }
    pe[kk] = e; po[kk] = oo;
  }
}

// ================= host: Alpert/Legendre filter construction (k=4) =================
static void poly_compose_lin(const double* p, int deg, double a, double b, double* out) {
  double res[8] = {0};
  for (int i = deg; i >= 0; --i) {
    double tmp[8] = {0};
    for (int j = 0; j < 7; ++j) { tmp[j] += res[j] * a; tmp[j + 1] += res[j] * b; }
    tmp[0] += p[i];
    for (int j = 0; j < 8; ++j) res[j] = tmp[j];
  }
  for (int j = 0; j < 8; ++j) out[j] = res[j];
}
static double proj_half(const double* a, const double* b, bool upper) {
  double prod[16] = {0};
  for (int i = 0; i < 8; ++i)
    for (int j = 0; j < 8; ++j) prod[i + j] += a[i] * b[j];
  double s = 0.0;
  for (int n = 0; n < 16; ++n) {
    double p = prod[n];
    if (fabs(p) < 1e-8) p = 0.0;
    double w = upper ? (1.0 - pow(0.5, n + 1)) : pow(0.5, n + 1);
    s += p * w / (double)(n + 1);
  }
  return s;
}
static double polyval8(const double* c, double x) {
  double r = 0.0, xp = 1.0;
  for (int i = 0; i < 8; ++i) { r += c[i] * xp; xp *= x; }
  return r;
}
static void compute_filters(Filt* F) {
  const int k = 4;
  const double leg[4][4] = {{1,0,0,0},{0,1,0,0},{-0.5,0,1.5,0},{0,-1.5,0,2.5}};
  double phi_c[4][8] = {{0}}, phi2_c[4][8] = {{0}};
  for (int ki = 0; ki < k; ++ki) {
    double p1[8], p2[8];
    poly_compose_lin(leg[ki], ki, -1.0, 2.0, p1);   // P_ki(2x-1)
    poly_compose_lin(leg[ki], ki, -1.0, 4.0, p2);   // P_ki(4x-1)
    double s1 = sqrt(2.0 * ki + 1.0);
    for (int j = 0; j < 8; ++j) { phi_c[ki][j] = s1 * p1[j]; phi2_c[ki][j] = sqrt(2.0) * s1 * p2[j]; }
  }
  double psi1[4][8] = {{0}}, psi2[4][8] = {{0}};
  for (int ki = 0; ki < k; ++ki) {
    for (int j = 0; j < 8; ++j) { psi1[ki][j] = phi2_c[ki][j]; psi2[ki][j] = 0.0; }
    for (int i = 0; i < k; ++i) {
      double p = proj_half(phi2_c[ki], phi_c[i], false);
      for (int j = 0; j < 8; ++j) { psi1[ki][j] -= p * phi_c[i][j]; psi2[ki][j] -= p * phi_c[i][j]; }
    }
    for (int jj = 0; jj < ki; ++jj) {
      double p = proj_half(phi2_c[ki], psi1[jj], false);
      for (int j = 0; j < 8; ++j) { psi1[ki][j] -= p * psi1[jj][j]; psi2[ki][j] -= p * psi2[jj][j]; }
    }
    double n1 = proj_half(psi1[ki], psi1[ki], false);
    double n2 = proj_half(psi2[ki], psi2[ki], true);
    double nrm = sqrt(n1 + n2);
    for (int j = 0; j < 8; ++j) {
      psi1[ki][j] /= nrm; psi2[ki][j] /= nrm;
      if (fabs(psi1[ki][j]) < 1e-8) psi1[ki][j] = 0.0;
      if (fabs(psi2[ki][j]) < 1e-8) psi2[ki][j] = 0.0;
    }
  }
  const double tq[4] = {-0.8611363115940526, -0.3399810435848563, 0.3399810435848563, 0.8611363115940526};
  const double wq[4] = { 0.3478548451374538,  0.6521451548625461, 0.6521451548625461, 0.3478548451374538};
  double H0[4][4], H1[4][4], G0[4][4], G1[4][4];
  double s2 = sqrt(2.0);
  for (int ki = 0; ki < k; ++ki)
    for (int kp = 0; kp < k; ++kp) {
      double h0 = 0, h1 = 0, g0 = 0, g1 = 0;
      for (int q = 0; q < 4; ++q) {
        double xm = (tq[q] + 1.0) / 2.0, wm = wq[q] / 2.0;
        double pk = polyval8(phi_c[kp], xm);
        double xa = xm / 2.0, xb = (xm + 1.0) / 2.0;
        h0 += wm * polyval8(phi_c[ki], xa) * pk;
        h1 += wm * polyval8(phi_c[ki], xb) * pk;
        double pa = (xa <= 0.5) ? polyval8(psi1[ki], xa) : polyval8(psi2[ki], xa);
        double pb = (xb <= 0.5) ? polyval8(psi1[ki], xb) : polyval8(psi2[ki], xb);
        g0 += wm * pa * pk;
        g1 += wm * pb * pk;
      }
      H0[ki][kp] = h0 / s2; H1[ki][kp] = h1 / s2; G0[ki][kp] = g0 / s2; G1[ki][kp] = g1 / s2;
    }
  for (int i = 0; i < 4; ++i)
    for (int j = 0; j < 4; ++j) {
      if (fabs(H0[i][j]) < 1e-8) H0[i][j] = 0; if (fabs(H1[i][j]) < 1e-8) H1[i][j] = 0;
      if (fabs(G0[i][j]) < 1e-8) G0[i][j] = 0; if (fabs(G1[i][j]) < 1e-8) G1[i][j] = 0;
    }
  for (int j = 0; j < 4; ++j)
    for (int kk = 0; kk < 4; ++kk) {
      F->ec_s[j][kk]     = (float)H0[kk][j];   // EC_S = [H0.T ; H1.T]
      F->ec_s[j + 4][kk] = (float)H1[kk][j];
      F->ec_d[j][kk]     = (float)G0[kk][j];   // EC_D = [G0.T ; G1.T]
      F->ec_d[j + 4][kk] = (float)G1[kk][j];
      F->rc_e[j][kk]     = (float)H0[j][kk];   // RC_E = [H0 ; G0]
      F->rc_e[j + 4][kk] = (float)G0[j][kk];
      F->rc_o[j][kk]     = (float)H1[j][kk];   // RC_O = [H1 ; G1]
      F->rc_o[j + 4][kk] = (float)G1[j][kk];
    }
}

// ================= orchestration =================
extern "C" void kernel_launch(void* const* d_in, const int* in_sizes, int n_in,
                              void* d_out, int out_size, void* d_ws, size_t ws_size,
                              hipStream_t stream) {
  const float* x    = (const float*)d_in[0];
  const float* wAre = (const float*)d_in[1];
  const float* wAim = (const float*)d_in[2];
  const float* wBre = (const float*)d_in[3];
  const float* wBim = (const float*)d_in[4];
  const float* wCre = (const float*)d_in[5];
  const float* wCim = (const float*)d_in[6];
  const float* t0w  = (const float*)d_in[7];
  const float* t0b  = (const float*)d_in[8];
  float* out = (float*)d_out;
  (void)in_sizes; (void)n_in; (void)out_size; (void)ws_size;

  Filt F;
  compute_filters(&F);

  // -------- workspace layout (floats) --------
  float* ws = (float*)d_ws;
  size_t o = 0;
  const size_t cap_half = (size_t)BB * (NN / 2) * CKD;       // 8.39M floats
  float* dbuf = ws + o; o += cap_half;
  float* xp0  = ws + o; o += cap_half;
  float* xp1  = ws + o; o += cap_half;
  const size_t ud_total = (size_t)BB * CKD * (NN - 1);       // 16.78M floats
  float* UdA  = ws + o; o += ud_total;
  float* UsA  = ws + o; o += ud_total;
  float* Fdre = ws + o; o += (size_t)MODES * BI;
  float* Fdim = ws + o; o += (size_t)MODES * BI;
  float* Fsre = ws + o; o += (size_t)MODES * BI;
  float* Fsim = ws + o; o += (size_t)MODES * BI;
  float* UdP  = ws + o; o += (size_t)2 * MODES * BI;
  float* UsP  = ws + o; o += (size_t)2 * MODES * BI;
  float* tcb  = ws + o; o += (size_t)MODES * (NN / 2);
  float* tsb  = ws + o; o += (size_t)MODES * (NN / 2);
  float* tib  = ws + o; o += (size_t)(NN / 2) * 2 * MODES;
  const size_t WSZ = (size_t)64 * 64 * 64;
  float* wT[6];
  for (int i = 0; i < 6; ++i) { wT[i] = ws + o; o += WSZ; }

  // -------- weight transposes (coalesced WMMA B loads) --------
  {
    const float* wsrc[6] = {wAre, wAim, wBre, wBim, wCre, wCim};
    for (int i = 0; i < 6; ++i)
      k_transpose_w<<<(int)((WSZ + 255) / 256), 256, 0, stream>>>(wsrc[i], wT[i]);
  }

  // Ud/Us per-level offsets
  size_t udoff[13];
  {
    size_t acc = 0; int Nl = NN;
    for (int i = 0; i < 13; ++i) { udoff[i] = acc; Nl >>= 1; acc += (size_t)BB * CKD * Nl; }
  }

  // -------- decomposition levels --------
  const float* xcur = x;
  float* pb[2] = {xp0, xp1};
  int pi = 0;
  int N = NN;
  for (int lev = 0; lev < 13; ++lev) {
    int Nh = N >> 1;
    int l = (Nh / 2 + 1 < 64) ? (Nh / 2 + 1) : 64;
    int lpad = ((l + 15) / 16) * 16;
    int K2 = ((2 * l + 3) / 4) * 4;
    float* xnext = pb[pi];

    int total = BB * Nh * CC;
    k_decompose<<<(total + 255) / 256, 256, 0, stream>>>(xcur, dbuf, xnext, Nh, F);

    int tf = lpad * Nh;
    k_twid_fwd<<<(tf + 255) / 256, 256, 0, stream>>>(tcb, tsb, Nh, l, lpad);
    int tv = Nh * K2;
    k_twid_inv<<<(tv + 255) / 256, 256, 0, stream>>>(tib, Nh, l, K2);

    if (Nh >= 32) {
      int tiles = (lpad / 16) * 128;
      k_dft_fwd_wmma<<<tiles / 4, 128, 0, stream>>>(dbuf,  tcb, tsb, Fdre, Fdim, Nh);
      k_dft_fwd_wmma<<<tiles / 4, 128, 0, stream>>>(xnext, tcb, tsb, Fsre, Fsim, Nh);
    } else {
      int tot = l * BI;
      k_dft_fwd_scalar<<<(tot + 255) / 256, 256, 0, stream>>>(dbuf,  tcb, tsb, Fdre, Fdim, Nh, l);
      k_dft_fwd_scalar<<<(tot + 255) / 256, 256, 0, stream>>>(xnext, tcb, tsb, Fsre, Fsim, Nh, l);
    }

    k_modemix_wmma<<<l, 256, 0, stream>>>(Fdre, Fdim, Fsre, Fsim,
                                          wT[0], wT[1], wT[2], wT[3], wT[4], wT[5],
                                          UdP, UsP);

    if (Nh >= 32) {
      int tiles = (Nh / 16) * 128;
      k_dft_inv_wmma<<<tiles / 4, 128, 0, stream>>>(UdP, tib, UdA + udoff[lev], Nh, K2);
      k_dft_inv_wmma<<<tiles / 4, 128, 0, stream>>>(UsP, tib, UsA + udoff[lev], Nh, K2);
    } else {
      int tot = Nh * BI;
      k_dft_inv_scalar<<<(tot + 255) / 256, 256, 0, stream>>>(UdP, tib, UdA + udoff[lev], Nh, K2);
      k_dft_inv_scalar<<<(tot + 255) / 256, 256, 0, stream>>>(UsP, tib, UsA + udoff[lev], Nh, K2);
    }

    xcur = xnext;
    pi ^= 1;
    N = Nh;
  }

  // -------- bottom transform --------
  {
    float* xnext = pb[pi];
    k_t0<<<(BB * CC * 4 + 255) / 256, 256, 0, stream>>>(xcur, t0w, t0b, xnext);
    xcur = xnext; pi ^= 1;
  }

  // -------- reconstruction levels --------
  for (int i = 12; i >= 0; --i) {
    int M = NN >> (i + 1);
    float* xnext = (i == 0) ? out : pb[pi];
    int total = BB * M * CC;
    k_recon<<<(total + 255) / 256, 256, 0, stream>>>(xcur, UsA + udoff[i], UdA + udoff[i], xnext, M, F);
    xcur = xnext; pi ^= 1;
  }
}